// KMaxPooling_36378372997288
// MI455X (gfx1250) — compile-verified
//
#include <hip/hip_runtime.h>
#include <hip/hip_bf16.h>

// ---------------------------------------------------------------------------
// k-max pooling (K=8) over S for x[B=32, S=4096, C=768] f32.
// Memory-bound: 402 MB read @ 23.3 TB/s => ~17 us floor. Strategy:
//   kernel 1: split S into 16 chunks; thread = (b, chunk, 4 channels);
//             float4 NT loads (coalesced 512B/wave), guarded top-8 insert in
//             registers; partial top-8s -> 12 MB workspace [B][16][8][C].
//   kernel 2: thread = (b, c); merge 16*8 candidates -> sorted top-8 -> out.
// Fallback single kernel if ws_size < 12 MB.
// ---------------------------------------------------------------------------

typedef __attribute__((ext_vector_type(4))) float fvec4;

namespace {
constexpr int Bn   = 32;
constexpr int Sn   = 4096;
constexpr int Cn   = 768;
constexpr int Kn   = 8;
constexpr int NCH  = 16;             // S chunks
constexpr int CHUNK = Sn / NCH;      // 256 rows per chunk
constexpr int T1   = Cn / 4;         // 192 threads, 4 channels each
}

// Insert v into sorted-descending t[0..7] (registers); cheap guard, rare bubble.
__device__ __forceinline__ void topk_insert(float v, float* t) {
  if (v > t[7]) {
    t[7] = v;
#pragma unroll
    for (int i = 7; i >= 1; --i) {
      float hi = fmaxf(t[i - 1], t[i]);
      float lo = fminf(t[i - 1], t[i]);
      t[i - 1] = hi;
      t[i] = lo;
    }
  }
}

// Kernel 1: per-chunk partial top-8. grid = (NCH, B), block = T1 (6 waves).
__global__ void kmax_partial(const float* __restrict__ x,
                             float* __restrict__ ws) {
  const int chunk = blockIdx.x;
  const int b     = blockIdx.y;
  const int c4    = threadIdx.x * 4;

  float ta[Kn], tb[Kn], tc[Kn], td[Kn];
#pragma unroll
  for (int i = 0; i < Kn; ++i) {
    ta[i] = -INFINITY; tb[i] = -INFINITY; tc[i] = -INFINITY; td[i] = -INFINITY;
  }

  const float* p =
      x + ((size_t)b * Sn + (size_t)chunk * CHUNK) * Cn + c4;

#pragma unroll 4
  for (int s = 0; s < CHUNK; ++s) {
    if ((s & 31) == 0) {
      // gfx1250 speculative prefetch (global_prefetch_b8): ~40 rows ahead
      // to cover HBM latency; OOB translation failures are silently dropped.
      __builtin_prefetch(p + (size_t)40 * Cn, 0, 0);
    }
    // Streamed exactly once: non-temporal b128 load keeps L2 for the ws/merge.
    const fvec4 v = __builtin_nontemporal_load((const fvec4*)p);
    p += Cn;
    topk_insert(v.x, ta);
    topk_insert(v.y, tb);
    topk_insert(v.z, tc);
    topk_insert(v.w, td);
  }

  // ws layout [B][NCH][Kn][C]: lane-consecutive float4 stores (coalesced).
  float* w = ws + (((size_t)b * NCH + chunk) * Kn) * Cn + c4;
#pragma unroll
  for (int i = 0; i < Kn; ++i) {
    fvec4 o; o.x = ta[i]; o.y = tb[i]; o.z = tc[i]; o.w = td[i];
    *(fvec4*)(w + (size_t)i * Cn) = o;
  }
}

// Kernel 2: merge NCH*Kn candidates per (b,c) -> sorted top-8 -> out[b][c*8+i].
__global__ void kmax_merge(const float* __restrict__ ws,
                           float* __restrict__ out) {
  const int idx = blockIdx.x * blockDim.x + threadIdx.x;  // 0 .. B*C-1
  const int b = idx / Cn;
  const int c = idx % Cn;

  float t[Kn];
#pragma unroll
  for (int i = 0; i < Kn; ++i) t[i] = -INFINITY;

  const float* p = ws + (size_t)b * NCH * Kn * Cn + c;
#pragma unroll 8
  for (int j = 0; j < NCH * Kn; ++j) {
    topk_insert(p[(size_t)j * Cn], t);
  }

  float* o = out + (size_t)idx * Kn;   // 32B-aligned (idx*8 floats)
  fvec4 lo; lo.x = t[0]; lo.y = t[1]; lo.z = t[2]; lo.w = t[3];
  fvec4 hi; hi.x = t[4]; hi.y = t[5]; hi.z = t[6]; hi.w = t[7];
  *(fvec4*)(o + 0) = lo;
  *(fvec4*)(o + 4) = hi;
}

// Fallback (ws too small): thread = (b,c), full-S scan.
__global__ void kmax_full(const float* __restrict__ x,
                          float* __restrict__ out) {
  const int idx = blockIdx.x * blockDim.x + threadIdx.x;  // 0 .. B*C-1
  const int b = idx / Cn;
  const int c = idx % Cn;

  float t[Kn];
#pragma unroll
  for (int i = 0; i < Kn; ++i) t[i] = -INFINITY;

  const float* p = x + (size_t)b * Sn * Cn + c;
#pragma unroll 4
  for (int s = 0; s < Sn; ++s) {
    if ((s & 31) == 0) __builtin_prefetch(p + (size_t)40 * Cn, 0, 0);
    topk_insert(__builtin_nontemporal_load(p), t);
    p += Cn;
  }

  float* o = out + (size_t)idx * Kn;
  fvec4 lo; lo.x = t[0]; lo.y = t[1]; lo.z = t[2]; lo.w = t[3];
  fvec4 hi; hi.x = t[4]; hi.y = t[5]; hi.z = t[6]; hi.w = t[7];
  *(fvec4*)(o + 0) = lo;
  *(fvec4*)(o + 4) = hi;
}

extern "C" void kernel_launch(void* const* d_in, const int* in_sizes, int n_in,
                              void* d_out, int out_size, void* d_ws, size_t ws_size,
                              hipStream_t stream) {
  (void)in_sizes; (void)n_in; (void)out_size;
  const float* x = (const float*)d_in[0];
  float* out = (float*)d_out;

  const size_t ws_need = (size_t)Bn * NCH * Kn * Cn * sizeof(float);  // 12 MB
  const int merge_blocks = (Bn * Cn) / 256;  // 96

  if (ws_size >= ws_need && d_ws != nullptr) {
    float* ws = (float*)d_ws;
    dim3 g1(NCH, Bn);                 // 512 blocks x 6 waves
    kmax_partial<<<g1, T1, 0, stream>>>(x, ws);
    kmax_merge<<<merge_blocks, 256, 0, stream>>>(ws, out);
  } else {
    kmax_full<<<merge_blocks, 256, 0, stream>>>(x, out);
  }
}